// DualGating_GNN_5858335391830
// MI455X (gfx1250) — compile-verified
//
#include <hip/hip_runtime.h>
#include <hip/hip_bf16.h>
#include <math.h>

#define N_NODES_C 100000
#define N_EDGES_C 1600000
#define NFEAT_C   128
#define NHID_C    64
#define NCLASS_C  40
#define NLAYERS_C 4

typedef __attribute__((ext_vector_type(2))) float v2f;
typedef __attribute__((ext_vector_type(8))) float v8f;

__device__ __forceinline__ v8f wmma_f32x4(v2f a, v2f b, v8f c) {
    // D = A(16x4 f32) * B(4x16 f32) + C(16x16 f32)
    return __builtin_amdgcn_wmma_f32_16x16x4_f32(
        /*neg_a=*/false, a, /*neg_b=*/false, b,
        /*c_mod=*/(short)0, c, /*reuse_a=*/false, /*reuse_b=*/false);
}

// OUT[M x ldo] = A[M x K] * W[Nvalid x K]^T (+bias) (optional relu)
// Optional epilogue: acc[m*NHID + n] = dinv[m]^2 * OUT_raw (for GCN self-loop init).
// One wave computes a 16-row x (NT*16)-col strip. M must be a multiple of 16.
template <int K, int NT, bool GUARD>
__global__ __launch_bounds__(256) void gemm_wmma(
    const float* __restrict__ A, const float* __restrict__ W,
    const float* __restrict__ bias, float* __restrict__ OUT,
    int M, int Nvalid, int ldo, int relu,
    const float* __restrict__ dinv, float* __restrict__ acc) {
    const int lane = threadIdx.x & 31;
    const int wave = threadIdx.x >> 5;
    const int m0 = (blockIdx.x * 8 + wave) * 16;
    if (m0 >= M) return;                 // wave-uniform: EXEC all-1s at WMMA

    const int half = lane >> 4;          // 0: lanes 0-15, 1: lanes 16-31
    const int l15  = lane & 15;
    const int mrow = m0 + l15;

    v8f c[NT];
#pragma unroll
    for (int t = 0; t < NT; ++t) { v8f z = {}; c[t] = z; }

    const float* Arow = A + (size_t)mrow * K + 2 * half;
#pragma unroll
    for (int k = 0; k < K; k += 4) {
        // A 16x4 f32 layout: lanes 0-15 hold K=k,k+1; lanes 16-31 hold K=k+2,k+3
        v2f a = *(const v2f*)(Arow + k);
#pragma unroll
        for (int t = 0; t < NT; ++t) {
            const int n = t * 16 + l15;
            v2f b;
            if (GUARD && n >= Nvalid) {
                b.x = 0.0f; b.y = 0.0f;
            } else {
                // B(k,n) = W[n][k]; mirror of A layout across lane halves
                b = *(const v2f*)(W + (size_t)n * K + k + 2 * half);
            }
            c[t] = wmma_f32x4(a, b, c[t]);
        }
    }

#pragma unroll
    for (int t = 0; t < NT; ++t) {
#pragma unroll
        for (int r = 0; r < 8; ++r) {
            const int mm = m0 + r + half * 8;   // C/D: VGPR r -> M=r (+8 for hi half)
            const int nn = t * 16 + l15;
            if (GUARD && nn >= Nvalid) continue;
            float v = c[t][r];
            if (bias) v += bias[nn];
            if (relu) v = fmaxf(v, 0.0f);
            OUT[(size_t)mm * ldo + nn] = v;
            if (acc) {
                const float di = dinv[mm];
                acc[(size_t)mm * NHID_C + nn] = di * di * v;
            }
        }
    }
}

__global__ __launch_bounds__(256) void k_init(float* deg, float* cnt, float* s, int n) {
    int i = blockIdx.x * blockDim.x + threadIdx.x;
    if (i < n) { deg[i] = 1.0f; cnt[i] = 0.0f; s[i] = 0.0f; }  // deg=1: self-loop
}

__global__ __launch_bounds__(256) void k_degree(const int* __restrict__ row,
                                                const int* __restrict__ col,
                                                float* deg, float* cnt, int E) {
    int e = blockIdx.x * blockDim.x + threadIdx.x;
    if (e < E) {
        atomicAdd(deg + col[e], 1.0f);   // in-degree over dst (incl. self-loop init)
        atomicAdd(cnt + row[e], 1.0f);   // gamma scatter-mean count over src
    }
}

__global__ __launch_bounds__(256) void k_dinv(const float* deg, const float* cnt,
                                              float* dinv, float* cinv, int n) {
    int i = blockIdx.x * blockDim.x + threadIdx.x;
    if (i < n) {
        dinv[i] = rsqrtf(deg[i]);                 // deg >= 1 always (self-loop)
        cinv[i] = 1.0f / fmaxf(cnt[i], 1.0f);
    }
}

// Fused per-edge pass: gamma |X[r]-X[c]|^2 reduction into s[r], and
// GCN scatter acc[c] += dinv[r]*dinv[c]*H[r]. One wave per edge, 2 feats/lane.
__global__ __launch_bounds__(256) void k_edge(
    const int* __restrict__ row, const int* __restrict__ col,
    const float* __restrict__ X, const float* __restrict__ H,
    const float* __restrict__ dinv,
    float* __restrict__ acc, float* __restrict__ s, int E) {
    const int lane = threadIdx.x & 31;
    int wave = (blockIdx.x * blockDim.x + threadIdx.x) >> 5;
    const int stride = (gridDim.x * blockDim.x) >> 5;
    for (int e = wave; e < E; e += stride) {
        const int r = row[e];
        const int c = col[e];
        const float2 xr = ((const float2*)(X + (size_t)r * NHID_C))[lane];
        const float2 xc = ((const float2*)(X + (size_t)c * NHID_C))[lane];
        const float2 hr = ((const float2*)(H + (size_t)r * NHID_C))[lane];
        const float dx = xr.x - xc.x;
        const float dy = xr.y - xc.y;
        float d = dx * dx + dy * dy;       // P == 2.0
#pragma unroll
        for (int off = 16; off > 0; off >>= 1) d += __shfl_xor(d, off, 32);
        if (lane == 0) atomicAdd(s + r, d);
        const float nrm = dinv[r] * dinv[c];
        float* ap = acc + (size_t)c * NHID_C + lane * 2;
        atomicAdd(ap, nrm * hr.x);
        atomicAdd(ap + 1, nrm * hr.y);
    }
}

// X = (X + g*relu(acc + conv_b) + g*skip) / (1 + 2g), g = tanh(s * cinv).
// Also zeroes s for the next layer. One wave per node.
__global__ __launch_bounds__(256) void k_combine(
    float* __restrict__ X, const float* __restrict__ acc,
    const float* __restrict__ skip, const float* __restrict__ bias,
    float* __restrict__ s, const float* __restrict__ cinv, int n) {
    const int t = blockIdx.x * blockDim.x + threadIdx.x;
    const int node = t >> 5;
    const int lane = t & 31;
    if (node >= n) return;
    const float g = tanhf(s[node] * cinv[node]);
    const float inv = 1.0f / (1.0f + 2.0f * g);
    const float2 av = ((const float2*)(acc + (size_t)node * NHID_C))[lane];
    const float2 bv = ((const float2*)bias)[lane];
    const float2 xv = ((const float2*)(X + (size_t)node * NHID_C))[lane];
    const float2 sk = ((const float2*)(skip + (size_t)node * NHID_C))[lane];
    const float ax = fmaxf(av.x + bv.x, 0.0f);
    const float ay = fmaxf(av.y + bv.y, 0.0f);
    float2 o;
    o.x = (xv.x + g * (ax + sk.x)) * inv;
    o.y = (xv.y + g * (ay + sk.y)) * inv;
    ((float2*)(X + (size_t)node * NHID_C))[lane] = o;
    if (lane == 0) s[node] = 0.0f;
}

extern "C" void kernel_launch(void* const* d_in, const int* in_sizes, int n_in,
                              void* d_out, int out_size, void* d_ws, size_t ws_size,
                              hipStream_t stream) {
    const float* x      = (const float*)d_in[0];
    const int*   eidx   = (const int*)d_in[1];
    const float* enc_W  = (const float*)d_in[2];
    const float* enc_b  = (const float*)d_in[3];
    const float* conv_W = (const float*)d_in[4];
    const float* conv_b = (const float*)d_in[5];
    const float* W_skip = (const float*)d_in[6];
    const float* dec_W  = (const float*)d_in[7];
    const float* dec_b  = (const float*)d_in[8];
    float* out = (float*)d_out;

    const int n = N_NODES_C;
    const int E = N_EDGES_C;
    const int* row = eidx;       // edge_index[0]
    const int* col = eidx + E;   // edge_index[1]

    float* ws = (float*)d_ws;
    float* X    = ws; ws += (size_t)n * NHID_C;
    float* skip = ws; ws += (size_t)n * NHID_C;
    float* H    = ws; ws += (size_t)n * NHID_C;
    float* acc  = ws; ws += (size_t)n * NHID_C;
    float* deg  = ws; ws += n;
    float* cnt  = ws; ws += n;
    float* dinv = ws; ws += n;
    float* cinv = ws; ws += n;
    float* s    = ws; ws += n;

    const dim3 blk(256);
    const int nb_n      = (n + 255) / 256;
    const int nb_e      = (E + 255) / 256;
    const int nb_gemm   = (n + 127) / 128;     // 8 waves * 16 rows per block
    const int nb_node32 = (int)(((size_t)n * 32 + 255) / 256);

    // Degree / norm factors (independent of X, computed once)
    k_init<<<nb_n, blk, 0, stream>>>(deg, cnt, s, n);
    k_degree<<<nb_e, blk, 0, stream>>>(row, col, deg, cnt, E);
    k_dinv<<<nb_n, blk, 0, stream>>>(deg, cnt, dinv, cinv, n);

    // Encoder: X = relu(x @ enc_W^T + enc_b)
    gemm_wmma<NFEAT_C, 4, false><<<nb_gemm, blk, 0, stream>>>(
        x, enc_W, enc_b, X, n, NHID_C, NHID_C, 1, nullptr, nullptr);
    // skip = X @ W_skip^T (constant across layers)
    gemm_wmma<NHID_C, 4, false><<<nb_gemm, blk, 0, stream>>>(
        X, W_skip, nullptr, skip, n, NHID_C, NHID_C, 0, nullptr, nullptr);

    for (int l = 0; l < NLAYERS_C; ++l) {
        // H = X @ conv_W^T; epilogue: acc = dinv^2 * H (self-loop term)
        gemm_wmma<NHID_C, 4, false><<<nb_gemm, blk, 0, stream>>>(
            X, conv_W, nullptr, H, n, NHID_C, NHID_C, 0, dinv, acc);
        // Fused gamma + GCN scatter over edges
        k_edge<<<4096, blk, 0, stream>>>(row, col, X, H, dinv, acc, s, E);
        // Gated combine (also resets s)
        k_combine<<<nb_node32, blk, 0, stream>>>(X, acc, skip, conv_b, s, cinv, n);
    }

    // Decoder: out = X @ dec_W^T + dec_b  (N = 40, guarded tiles)
    gemm_wmma<NHID_C, 3, true><<<nb_gemm, blk, 0, stream>>>(
        X, dec_W, dec_b, out, n, NCLASS_C, NCLASS_C, 0, nullptr, nullptr);
}